// INRF_20194936226279
// MI455X (gfx1250) — compile-verified
//
#include <hip/hip_runtime.h>
#include <hip/hip_bf16.h>
#include <math.h>

typedef __attribute__((ext_vector_type(2))) float v2f;
typedef __attribute__((ext_vector_type(4))) float v4f;
typedef __attribute__((ext_vector_type(8))) float v8f;
typedef __attribute__((ext_vector_type(4))) int   v4i;

#define BB   4
#define HH   32
#define WW   32
#define CC   8
#define FF   8
#define KKH  5
#define KKW  5
#define KTOT (KKH * KKW * CC)   // 200 = 50 chunks of 4
#define NPIX (HH * WW)          // 1024 pixels per batch
#define NFLAT (NPIX * CC)       // 8192 per batch
#define TOTAL (BB * NFLAT)      // 32768

// ---------------- hardware tanh (gfx1250 V_TANH_F32) ----------------
__device__ __forceinline__ float tanh_hw(float x) {
#if __has_builtin(__builtin_amdgcn_tanhf)
    return __builtin_amdgcn_tanhf(x);
#elif __has_builtin(__builtin_amdgcn_tanh_f32)
    return __builtin_amdgcn_tanh_f32(x);
#else
    return tanhf(x);
#endif
}

// Zero-padded NHWC fetch: clamped address + select (no divergent branches;
// lowers to uniform global_load + v_cndmask instead of s_and_saveexec).
__device__ __forceinline__ float fetch_im(const float* __restrict__ img,
                                          int b, int h, int w, int c) {
    const bool inb = (h >= 0) & (h < HH) & (w >= 0) & (w < WW);
    const int hc = min(max(h, 0), HH - 1);
    const int wc = min(max(w, 0), WW - 1);
    const float v = img[(((b * HH) + hc) * WW + wc) * CC + c];
    return inb ? v : 0.0f;
}

// ======================================================================
// Kernel 1: fused conv(inputs,M) -> first  and conv(inputs,G) -> blurred
// One wave per 16-pixel tile; implicit GEMM via V_WMMA_F32_16X16X4_F32.
// A (16x4 f32): lanes 0-15 hold K={k,k+1}, lanes 16-31 hold K={k+2,k+3}.
// D (16x16 f32, v8f): lanes 0-15 rows 0-7, lanes 16-31 rows 8-15, N=lane%16.
// ======================================================================
__global__ __launch_bounds__(32)
void inrf_conv_mg(const float* __restrict__ in, const float* __restrict__ Mk,
                  const float* __restrict__ Gk, float* __restrict__ first,
                  float* __restrict__ blur) {
    const int tile = blockIdx.x;      // 256 tiles of 16 pixels
    const int pix0 = tile << 4;       // flat pixel over (b,h,w)
    const int b    = pix0 >> 10;
    const int rem  = pix0 & 1023;
    const int hrow = rem >> 5;
    const int w0   = rem & 31;        // multiple of 16 (WW=32)
    const int lane = threadIdx.x;
    const int g    = lane >> 4;       // K-pair group
    const int mr   = lane & 15;       // A-row / B-col index
    const int wpix = w0 + mr;         // this lane's A-row pixel (w coord)
    const int fcol = mr & 7;          // filter column (only mr<8 stored)

    v8f accM = {};
    v8f accG = {};
    #pragma unroll 2
    for (int kc = 0; kc < KTOT / 4; ++kc) {
        const int k0  = kc * 4 + 2 * g;
        const int k1  = k0 + 1;
        const int kh0 = k0 / 40, r0 = k0 - kh0 * 40;
        const int kh1 = k1 / 40, r1 = k1 - kh1 * 40;
        v2f A;
        A.x = fetch_im(in, b, hrow + kh0 - 2, wpix + (r0 >> 3) - 2, r0 & 7);
        A.y = fetch_im(in, b, hrow + kh1 - 2, wpix + (r1 >> 3) - 2, r1 & 7);
        v2f Bm = { Mk[k0 * FF + fcol], Mk[k1 * FF + fcol] };
        v2f Bg = { Gk[k0 * FF + fcol], Gk[k1 * FF + fcol] };
        accM = __builtin_amdgcn_wmma_f32_16x16x4_f32(false, A, false, Bm,
                                                     (short)0, accM, false, false);
        accG = __builtin_amdgcn_wmma_f32_16x16x4_f32(false, A, false, Bg,
                                                     (short)0, accG, false, false);
    }
    if (mr < FF) {
        #pragma unroll
        for (int r = 0; r < 8; ++r) {
            const int p = pix0 + g * 8 + r;   // D row = pixel within tile
            first[p * FF + mr] = accM[r];
            blur[p * FF + mr]  = accG[r];
        }
    }
}

// ======================================================================
// Kernel 2: partial[pc][b,q] = sum_{p in chunk pc} tanh(x[b,p] - blur[b,q])
// 512 blocks = 4 batches x 32 q-chunks(256) x 4 p-chunks(2048).
// p-chunk staged into LDS via GLOBAL_LOAD_ASYNC_TO_LDS_B128 (ASYNCcnt path);
// broadcast LDS reads; fixed-order sums (bitwise deterministic).
// ======================================================================
#define PCHUNK 2048
#define NPCH   4
#define QPB    256

__global__ __launch_bounds__(256)
void inrf_pairwise(const float* __restrict__ x, const float* __restrict__ qv,
                   float* __restrict__ partial) {
    __shared__ float xs[PCHUNK];
    const int id  = blockIdx.x;
    const int pc  = id & 3;
    const int qc  = (id >> 2) & 31;
    const int b   = id >> 7;
    const int tid = threadIdx.x;

    const float* xb = x + b * NFLAT + pc * PCHUNK;

#if __has_builtin(__builtin_amdgcn_global_load_async_to_lds_b128)
    {
        typedef __attribute__((address_space(1))) v4i* gv4p;  // global int4*
        typedef __attribute__((address_space(3))) v4i* lv4p;  // LDS int4*
        gv4p gsrc = (gv4p)(uintptr_t)xb;
        lv4p ldst = (lv4p)(uint32_t)(uintptr_t)xs;
        // 512 x 16B transfers: two per thread (direct global->LDS, no VGPRs)
        __builtin_amdgcn_global_load_async_to_lds_b128(gsrc + tid,       ldst + tid,       0, 0);
        __builtin_amdgcn_global_load_async_to_lds_b128(gsrc + tid + 256, ldst + tid + 256, 0, 0);
    }
  #if __has_builtin(__builtin_amdgcn_s_wait_asynccnt)
    __builtin_amdgcn_s_wait_asynccnt(0);
  #else
    asm volatile("s_wait_asynccnt 0x0" ::: "memory");
  #endif
#else
    for (int i = tid; i < PCHUNK; i += 256) xs[i] = xb[i];
#endif
    __syncthreads();

    const int   qi = b * NFLAT + qc * QPB + tid;
    const float q  = qv[qi];
    float a0 = 0.f, a1 = 0.f, a2 = 0.f, a3 = 0.f;
    #pragma unroll 4
    for (int p = 0; p < PCHUNK; p += 4) {
        a0 += tanh_hw(xs[p + 0] - q);
        a1 += tanh_hw(xs[p + 1] - q);
        a2 += tanh_hw(xs[p + 2] - q);
        a3 += tanh_hw(xs[p + 3] - q);
    }
    partial[pc * TOTAL + qi] = (a0 + a1) + (a2 + a3);
}

// Kernel 2b: deterministic fixed-order reduction of the 4 p-chunk partials.
__global__ __launch_bounds__(256)
void inrf_reduce(const float* __restrict__ partial, float* __restrict__ diff) {
    const int i = blockIdx.x * 256 + threadIdx.x;
    if (i < TOTAL) {
        diff[i] = ((partial[i] + partial[TOTAL + i]) +
                   (partial[2 * TOTAL + i] + partial[3 * TOTAL + i]));
    }
}

// ======================================================================
// Kernel 3: out = first - conv5x5(diff, W)   (same WMMA tiling as kernel 1)
// ======================================================================
__global__ __launch_bounds__(32)
void inrf_conv_w(const float* __restrict__ dif, const float* __restrict__ Wk,
                 const float* __restrict__ first, float* __restrict__ out) {
    const int tile = blockIdx.x;
    const int pix0 = tile << 4;
    const int b    = pix0 >> 10;
    const int rem  = pix0 & 1023;
    const int hrow = rem >> 5;
    const int w0   = rem & 31;
    const int lane = threadIdx.x;
    const int g    = lane >> 4;
    const int mr   = lane & 15;
    const int wpix = w0 + mr;
    const int fcol = mr & 7;

    v8f acc = {};
    #pragma unroll 2
    for (int kc = 0; kc < KTOT / 4; ++kc) {
        const int k0  = kc * 4 + 2 * g;
        const int k1  = k0 + 1;
        const int kh0 = k0 / 40, r0 = k0 - kh0 * 40;
        const int kh1 = k1 / 40, r1 = k1 - kh1 * 40;
        v2f A;
        A.x = fetch_im(dif, b, hrow + kh0 - 2, wpix + (r0 >> 3) - 2, r0 & 7);
        A.y = fetch_im(dif, b, hrow + kh1 - 2, wpix + (r1 >> 3) - 2, r1 & 7);
        v2f Bw = { Wk[k0 * FF + fcol], Wk[k1 * FF + fcol] };
        acc = __builtin_amdgcn_wmma_f32_16x16x4_f32(false, A, false, Bw,
                                                    (short)0, acc, false, false);
    }
    if (mr < FF) {
        #pragma unroll
        for (int r = 0; r < 8; ++r) {
            const int p = pix0 + g * 8 + r;
            out[p * FF + mr] = first[p * FF + mr] - acc[r];
        }
    }
}

// ======================================================================
extern "C" void kernel_launch(void* const* d_in, const int* in_sizes, int n_in,
                              void* d_out, int out_size, void* d_ws, size_t ws_size,
                              hipStream_t stream) {
    const float* in = (const float*)d_in[0];   // (4,32,32,8)
    const float* Mk = (const float*)d_in[1];   // (5,5,8,8)
    const float* Wk = (const float*)d_in[2];   // (5,5,8,8)
    const float* Gk = (const float*)d_in[3];   // (5,5,8,8)
    float* out = (float*)d_out;

    float* ws      = (float*)d_ws;
    float* first   = ws;                 // 32768
    float* blur    = ws + TOTAL;         // 32768
    float* diff    = ws + 2 * TOTAL;     // 32768
    float* partial = ws + 3 * TOTAL;     // 4 * 32768

    inrf_conv_mg<<<(BB * NPIX) / 16, 32, 0, stream>>>(in, Mk, Gk, first, blur);
    inrf_pairwise<<<BB * (NFLAT / QPB) * NPCH, 256, 0, stream>>>(in, blur, partial);
    inrf_reduce<<<(TOTAL + 255) / 256, 256, 0, stream>>>(partial, diff);
    inrf_conv_w<<<(BB * NPIX) / 16, 32, 0, stream>>>(diff, Wk, first, out);
}